// UniDirAttentionItr_4217657885105
// MI455X (gfx1250) — compile-verified
//
#include <hip/hip_runtime.h>
#include <hip/hip_bf16.h>
#include <math.h>

#define BB 32
#define SS 8192
#define DD 256
#define NSCHUNK 8   // s-chunks for the pooling pass (partials, no atomics)

typedef float v2f __attribute__((ext_vector_type(2)));
typedef float v4f __attribute__((ext_vector_type(4)));
typedef float v8f __attribute__((ext_vector_type(8)));

// ---------------------------------------------------------------------------
// K1: scores[b,s] = seq[b,s,:] . vec[b,:]  via V_WMMA_F32_16X16X4_F32
// One wave computes 16 scores (M = 16 s-rows, K loops over D in steps of 8
// feeding TWO independent accumulators to break the XDL C-operand RAW chain).
// B operand = vector chunk broadcast to all 16 columns -> every column of the
// 16x16 accumulator holds the same 16 scores.
// ---------------------------------------------------------------------------
__global__ void __launch_bounds__(128)
k_scores(const float* __restrict__ seq, const float* __restrict__ vec,
         float* __restrict__ scores) {
    const int tilesPerB = SS / 64;               // 64 s per block (4 waves x 16)
    const int blk  = blockIdx.x;
    const int b    = blk / tilesPerB;
    const int tile = blk % tilesPerB;
    const int wave = threadIdx.x >> 5;
    const int lane = threadIdx.x & 31;
    const int lr   = lane & 15;                  // M row (A) / N col (B,C)
    const int lh   = lane >> 4;                  // K-half selector

    const int s0 = tile * 64 + wave * 16;
    const float* __restrict__ arow = seq + ((size_t)b * SS + (size_t)(s0 + lr)) * DD;
    const float* __restrict__ vrow = vec + (size_t)b * DD;

    v8f c0 = {}, c1 = {};
#pragma unroll 4
    for (int d0 = 0; d0 < DD; d0 += 8) {
        // A: lane holds A[M=lr, K = 2*lh + {0,1}]  (ISA 32-bit 16x4 A layout)
        v2f a0 = *reinterpret_cast<const v2f*>(arow + d0 + 2 * lh);
        v2f a1 = *reinterpret_cast<const v2f*>(arow + d0 + 4 + 2 * lh);
        // B: lane holds B[K = 2*lh + {0,1}, N=lr]; broadcast vector -> all N equal
        v2f b0 = *reinterpret_cast<const v2f*>(vrow + d0 + 2 * lh);
        v2f b1 = *reinterpret_cast<const v2f*>(vrow + d0 + 4 + 2 * lh);
        c0 = __builtin_amdgcn_wmma_f32_16x16x4_f32(false, a0, false, b0,
                                                   (short)0, c0, false, false);
        c1 = __builtin_amdgcn_wmma_f32_16x16x4_f32(false, a1, false, b1,
                                                   (short)0, c1, false, false);
    }
    v8f c = c0 + c1;

    // C/D layout: lane holds D[M = g + 8*lh, N = lr] in c[g]; columns identical.
    if (lr == 0) {
        float* out = scores + (size_t)b * SS + s0 + 8 * lh;
#pragma unroll
        for (int g = 0; g < 8; ++g) out[g] = c[g];
    }
}

// ---------------------------------------------------------------------------
// K2: stable masked softmax over S per batch row (tiny: 1 MB of scores).
// attn = exp(scores - max) * mask / (sum + 1e-13)
// ---------------------------------------------------------------------------
__global__ void __launch_bounds__(256)
k_softmax(const float* __restrict__ scores, const float* __restrict__ mask,
          float* __restrict__ attn) {
    __shared__ float red[256];
    const int b = blockIdx.x;
    const int t = threadIdx.x;
    const float* __restrict__ srow = scores + (size_t)b * SS;
    const float* __restrict__ mrow = mask   + (size_t)b * SS;
    float* __restrict__ arow       = attn   + (size_t)b * SS;

    float lm = -3.402823466e38f;
    for (int s = t; s < SS; s += 256) lm = fmaxf(lm, srow[s]);
    red[t] = lm; __syncthreads();
    for (int o = 128; o > 0; o >>= 1) {
        if (t < o) red[t] = fmaxf(red[t], red[t + o]);
        __syncthreads();
    }
    const float m = red[0];
    __syncthreads();

    float ls = 0.0f;
    for (int s = t; s < SS; s += 256) {
        float e = __expf(srow[s] - m) * mrow[s];
        arow[s] = e;
        ls += e;
    }
    red[t] = ls; __syncthreads();
    for (int o = 128; o > 0; o >>= 1) {
        if (t < o) red[t] += red[t + o];
        __syncthreads();
    }
    const float inv = 1.0f / (red[0] + 1e-13f);
    for (int s = t; s < SS; s += 256) arow[s] *= inv;
}

// ---------------------------------------------------------------------------
// K3: partial pooled: partial[b,chunk,d] = sum_{s in chunk} attn[b,s]*seq[b,s,d]
// A = attn broadcast across all M rows (D replicated per row), B = 4x16 seq
// tile. Two independent accumulators over s0 / s0+4; one b128 attn fetch
// feeds both WMMAs. One block = (b, chunk); 16 waves, one d-stripe per wave.
// ---------------------------------------------------------------------------
__global__ void __launch_bounds__(512)
k_pool(const float* __restrict__ seq, const float* __restrict__ attn,
       float* __restrict__ partial) {
    const int blk = blockIdx.x;
    const int b   = blk / NSCHUNK;
    const int sc  = blk % NSCHUNK;
    const int wave = threadIdx.x >> 5;
    const int lane = threadIdx.x & 31;
    const int lr   = lane & 15;
    const int lh   = lane >> 4;
    const int d0   = wave * 16;                  // 16 waves cover D=256

    const int schunk = SS / NSCHUNK;             // 1024
    const int sbeg   = sc * schunk;
    const float* __restrict__ sq = seq  + (size_t)b * SS * DD;
    const float* __restrict__ at = attn + (size_t)b * SS;

    v8f c0 = {}, c1 = {};
#pragma unroll 2
    for (int s0 = sbeg; s0 < sbeg + schunk; s0 += 8) {
        // attn[s0 + 2*lh .. s0 + 2*lh + 5] would span both chunks; fetch the
        // aligned quad covering this lane's K-pairs for both WMMAs.
        v2f a0, a1;
        {
            // lanes lh==0 need {s0,s0+1,s0+4,s0+5}; lh==1 need {s0+2,s0+3,s0+6,s0+7}
            v4f q0 = *reinterpret_cast<const v4f*>(at + s0 + 4 * lh);       // s0+4lh..+3
            v4f q1 = *reinterpret_cast<const v4f*>(at + s0 + 4 - 4 * lh);   // other quad
            // lh==0: a0 = q0.xy (s0,s0+1),   a1 = q1.xy (s0+4,s0+5)
            // lh==1: a0 = q0.zw? no -> a0 needs (s0+2,s0+3) = q1.zw when lh==1
            if (lh == 0) { a0.x = q0.x; a0.y = q0.y; a1.x = q1.x; a1.y = q1.y; }
            else         { a0.x = q1.z; a0.y = q1.w; a1.x = q0.z; a1.y = q0.w; }
        }
        // B[k,n] = seq[s0+k, d0+n]: lane holds K = 2*lh+{0,1}, N = lr
        v2f b0, b1;
        b0.x = sq[(size_t)(s0 + 2 * lh)         * DD + d0 + lr];
        b0.y = sq[(size_t)(s0 + 2 * lh + 1)     * DD + d0 + lr];
        b1.x = sq[(size_t)(s0 + 4 + 2 * lh)     * DD + d0 + lr];
        b1.y = sq[(size_t)(s0 + 4 + 2 * lh + 1) * DD + d0 + lr];
        // prefetch next iteration's rows (strided by D) -> global_prefetch_b8
        __builtin_prefetch(sq + (size_t)(s0 + 8 + 2 * lh) * DD + d0 + lr, 0, 0);
        c0 = __builtin_amdgcn_wmma_f32_16x16x4_f32(false, a0, false, b0,
                                                   (short)0, c0, false, false);
        c1 = __builtin_amdgcn_wmma_f32_16x16x4_f32(false, a1, false, b1,
                                                   (short)0, c1, false, false);
    }

    // All rows equal; c[0] on lane l is column N = l&15. Lanes 16..31 duplicate.
    if (lh == 0)
        partial[((size_t)(b * NSCHUNK + sc)) * DD + d0 + lr] = c0[0] + c1[0];
}

// ---------------------------------------------------------------------------
// K4: fold the NSCHUNK partials -> pooled[b,d]  (fixed order => deterministic)
// ---------------------------------------------------------------------------
__global__ void __launch_bounds__(256)
k_reduce(const float* __restrict__ partial, float* __restrict__ pooled) {
    const int i = blockIdx.x * blockDim.x + threadIdx.x;
    if (i >= BB * DD) return;
    const int b = i / DD;
    const int d = i % DD;
    float s = 0.0f;
#pragma unroll
    for (int cgroup = 0; cgroup < NSCHUNK; ++cgroup)
        s += partial[((size_t)(b * NSCHUNK + cgroup)) * DD + d];
    pooled[i] = s;
}

extern "C" void kernel_launch(void* const* d_in, const int* in_sizes, int n_in,
                              void* d_out, int out_size, void* d_ws, size_t ws_size,
                              hipStream_t stream) {
    (void)in_sizes; (void)n_in; (void)out_size; (void)ws_size;
    const float* seq  = (const float*)d_in[0];   // [B,S,D] fp32
    const float* vec  = (const float*)d_in[1];   // [B,D]   fp32
    const float* mask = (const float*)d_in[2];   // [B,S]   fp32

    float* pooled = (float*)d_out;               // [B,D]
    float* attn   = pooled + (size_t)BB * DD;    // [B,S]

    float* scores  = (float*)d_ws;                       // B*S floats (1 MB)
    float* partial = scores + (size_t)BB * SS;           // B*NSCHUNK*D floats

    // Pass 1: scores (reads sequence once, 268 MB)
    k_scores<<<BB * (SS / 64), 128, 0, stream>>>(seq, vec, scores);
    // Pass 2: masked softmax -> attn (tiny)
    k_softmax<<<BB, 256, 0, stream>>>(scores, mask, attn);
    // Pass 3: weighted pooling partials (reads sequence once, 268 MB)
    k_pool<<<BB * NSCHUNK, 512, 0, stream>>>(seq, attn, partial);
    // Pass 4: fold partials -> pooled
    k_reduce<<<(BB * DD + 255) / 256, 256, 0, stream>>>(partial, pooled);
}